// Network_75960791597064
// MI455X (gfx1250) — compile-verified
//
#include <hip/hip_runtime.h>
#include <hip/hip_bf16.h>
#include <cmath>

typedef __attribute__((ext_vector_type(2))) float v2f;
typedef __attribute__((ext_vector_type(8))) float v8f;

#define MULW 32
#define HID 64

__device__ __forceinline__ v8f wmma4(v2f a, v2f b, v8f c) {
  // V_WMMA_F32_16X16X4_F32 : D = A(16x4) x B(4x16) + C(16x16), all f32
  return __builtin_amdgcn_wmma_f32_16x16x4_f32(false, a, false, b, (short)0, c,
                                               false, false);
}

__device__ __forceinline__ float sigm(float x) { return 1.f / (1.f + __expf(-x)); }

__constant__ float kNull = 0.f; // unused; keeps file self-contained

#define INV_SQRT128 0.08838834764831845f
#define INV_SQRT256 0.0625f
#define INV_SQRT10  0.31622776601683794f
#define INV_SQRT64  0.125f
#define INV_SQRTNN  0.17677669529663687f /* 1/sqrt(32) */
#define INV_SQRT3F  0.5773502691896258f
#define SQRT3F      1.7320508075688772f
#define C_S 0.3826834323650898f  /* sin(pi/8) */
#define C_X 0.9238795325112867f  /* cos(pi/8) */
#define PI_F 3.14159265358979323846f
#define RMAXF 4.0f

// ---------------------------------------------------------------- utilities
__global__ void zero_kernel(float* __restrict__ p, long n) {
  long i = (long)blockIdx.x * blockDim.x + threadIdx.x;
  if (i < n) p[i] = 0.f;
}

__global__ void init_sv_kernel(const float* __restrict__ x, float* __restrict__ s,
                               float* __restrict__ v, int N) {
  long i = (long)blockIdx.x * blockDim.x + threadIdx.x;
  if (i >= (long)N * 128) return;
  int n = (int)(i >> 7), c = (int)(i & 127);
  float val = x[i];
  if (c < 32) s[n * 32 + c] = val;
  else        v[n * 96 + (c - 32)] = val;
}

__global__ void edge_pre_kernel(const float* __restrict__ ev, float* __restrict__ Y1,
                                float* __restrict__ emb, int E) {
  int e = blockIdx.x * blockDim.x + threadIdx.x;
  if (e >= E) return;
  float x0 = ev[e * 3 + 0], x1 = ev[e * 3 + 1], x2 = ev[e * 3 + 2];
  float len = sqrtf(x0 * x0 + x1 * x1 + x2 * x2);
  float inv = SQRT3F / (len + 1e-9f);
  Y1[e * 3 + 0] = x0 * inv;
  Y1[e * 3 + 1] = x1 * inv;
  Y1[e * 3 + 2] = x2 * inv;
  float xr = len / RMAXF;
  float u = 2.f * (xr - 1.f);
  float cut = (u > 0.f) ? 0.f : ((u < -1.f) ? 1.f : (1.f - __cosf(PI_F * u)) * 0.5f);
#pragma unroll
  for (int j = 0; j < 10; j++) {
    float center = (RMAXF / 9.0f) * (float)j;
    float t = (len - center) / (RMAXF / 10.0f);
    emb[e * 12 + j] = __expf(-t * t) * cut;
  }
  emb[e * 12 + 10] = 0.f;
  emb[e * 12 + 11] = 0.f;
}

// ------------------------------------------- node FCTP: (s,v) x attr x W -> out
// out_sa = fctp_s(s, W0a), out_sb = fctp_s(s, W0b)
// out_va = fctp_v(v, W1a), out_vb = fctp_v(v, W1b)   (norm 1/sqrt(128))
__global__ void node_fctp_kernel(const float* __restrict__ s, const float* __restrict__ v,
                                 const float* __restrict__ attr,
                                 const float* __restrict__ W0a, const float* __restrict__ W1a,
                                 const float* __restrict__ W0b, const float* __restrict__ W1b,
                                 float* __restrict__ out_sa, float* __restrict__ out_va,
                                 float* __restrict__ out_sb, float* __restrict__ out_vb,
                                 int ntiles, int N) {
  int wave = (int)(((long)blockIdx.x * blockDim.x + threadIdx.x) >> 5);
  if (wave >= ntiles) return;
  int lane = threadIdx.x & 31;
  int half = lane >> 4, lr = lane & 15;
  int node = wave * 16 + lr;
  if (node > N - 1) node = N - 1;
  float at0 = attr[node * 4 + half * 2];
  float at1 = attr[node * 4 + half * 2 + 1];

  { // scalar path, K = 128
    v8f acc[2][2] = {};
    for (int k0 = 0; k0 < 128; k0 += 4) {
      int u = k0 >> 2;
      int kr = k0 + half * 2;
      float sv = s[node * 32 + u];
      v2f a; a.x = sv * at0; a.y = sv * at1;
#pragma unroll
      for (int t = 0; t < 2; t++) {
        v2f b;
        b.x = W0a[kr * 32 + t * 16 + lr];
        b.y = W0a[(kr + 1) * 32 + t * 16 + lr];
        acc[0][t] = wmma4(a, b, acc[0][t]);
        b.x = W0b[kr * 32 + t * 16 + lr];
        b.y = W0b[(kr + 1) * 32 + t * 16 + lr];
        acc[1][t] = wmma4(a, b, acc[1][t]);
      }
    }
#pragma unroll
    for (int t = 0; t < 2; t++)
#pragma unroll
      for (int r = 0; r < 8; r++) {
        int row = wave * 16 + r + half * 8;
        if (row < N) {
          out_sa[row * 32 + t * 16 + lr] = acc[0][t][r] * INV_SQRT128;
          out_sb[row * 32 + t * 16 + lr] = acc[1][t][r] * INV_SQRT128;
        }
      }
  }

  { // vector path, 3 components, K = 128, B fragments shared across comps
    v8f acc[3][2][2] = {};
    for (int k0 = 0; k0 < 128; k0 += 4) {
      int u = k0 >> 2;
      int kr = k0 + half * 2;
      v2f bA[2], bB[2];
#pragma unroll
      for (int t = 0; t < 2; t++) {
        bA[t].x = W1a[kr * 32 + t * 16 + lr];
        bA[t].y = W1a[(kr + 1) * 32 + t * 16 + lr];
        bB[t].x = W1b[kr * 32 + t * 16 + lr];
        bB[t].y = W1b[(kr + 1) * 32 + t * 16 + lr];
      }
#pragma unroll
      for (int c = 0; c < 3; c++) {
        float vv = v[node * 96 + u * 3 + c];
        v2f a; a.x = vv * at0; a.y = vv * at1;
#pragma unroll
        for (int t = 0; t < 2; t++) {
          acc[c][0][t] = wmma4(a, bA[t], acc[c][0][t]);
          acc[c][1][t] = wmma4(a, bB[t], acc[c][1][t]);
        }
      }
    }
#pragma unroll
    for (int c = 0; c < 3; c++)
#pragma unroll
      for (int t = 0; t < 2; t++)
#pragma unroll
        for (int r = 0; r < 8; r++) {
          int row = wave * 16 + r + half * 8;
          if (row < N) {
            out_va[row * 96 + (t * 16 + lr) * 3 + c] = acc[c][0][t][r] * INV_SQRT128;
            out_vb[row * 96 + (t * 16 + lr) * 3 + c] = acc[c][1][t][r] * INV_SQRT128;
          }
        }
  }
}

// --------------------------------------------------------- edge MLP + scatter
#define EWAVES 4
__global__ void edge_kernel(const float* __restrict__ emb, const float* __restrict__ Y1,
                            const int* __restrict__ esrc, const int* __restrict__ edst,
                            const float* __restrict__ xs, const float* __restrict__ xv,
                            const float* __restrict__ Wfc1, const float* __restrict__ Wfc2,
                            float* __restrict__ aggs, float* __restrict__ aggv,
                            int E, int ntiles) {
  __shared__ float lds_h[EWAVES][16 * HID];
  __shared__ float lds_w[EWAVES][16 * 128];
  int wv = threadIdx.x >> 5;
  int wave = blockIdx.x * EWAVES + wv;
  int lane = threadIdx.x & 31;
  int half = lane >> 4, lr = lane & 15;
  bool active = (wave < ntiles);
  int tile = active ? wave : 0;
  int e0 = tile * 16;
  int eA = e0 + lr;
  if (eA > E - 1) eA = E - 1;

  // stage 1: h = silu(emb @ Wfc1 / sqrt(10)),  K padded 10->12
  v8f hacc[4] = {};
#pragma unroll
  for (int k0 = 0; k0 < 12; k0 += 4) {
    int kr = k0 + half * 2;
    v2f a; a.x = emb[eA * 12 + kr]; a.y = emb[eA * 12 + kr + 1];
#pragma unroll
    for (int t = 0; t < 4; t++) {
      v2f b;
      b.x = (kr < 10)     ? Wfc1[kr * HID + t * 16 + lr]       : 0.f;
      b.y = (kr + 1 < 10) ? Wfc1[(kr + 1) * HID + t * 16 + lr] : 0.f;
      hacc[t] = wmma4(a, b, hacc[t]);
    }
  }
#pragma unroll
  for (int t = 0; t < 4; t++)
#pragma unroll
    for (int r = 0; r < 8; r++) {
      float hx = hacc[t][r] * INV_SQRT10;
      lds_h[wv][(r + half * 8) * HID + t * 16 + lr] = hx * sigm(hx);
    }
  __syncthreads();

  // stage 2: w = h @ Wfc2 / sqrt(64),  K = 64, 8 output tiles
  v8f wacc[8] = {};
  for (int k0 = 0; k0 < HID; k0 += 4) {
    int kr = k0 + half * 2;
    v2f a; a.x = lds_h[wv][lr * HID + kr]; a.y = lds_h[wv][lr * HID + kr + 1];
#pragma unroll
    for (int t = 0; t < 8; t++) {
      v2f b;
      b.x = Wfc2[kr * 128 + t * 16 + lr];
      b.y = Wfc2[(kr + 1) * 128 + t * 16 + lr];
      wacc[t] = wmma4(a, b, wacc[t]);
    }
  }
#pragma unroll
  for (int t = 0; t < 8; t++)
#pragma unroll
    for (int r = 0; r < 8; r++)
      lds_w[wv][(r + half * 8) * 128 + t * 16 + lr] = wacc[t][r] * INV_SQRT64;
  __syncthreads();

  // stage 3: gather source features, form messages, atomic segment-sum
  for (int p = lane; p < 512; p += 32) {
    int el = p >> 5, m = p & 31;
    int e = e0 + el;
    if (active && e < E) {
      int src = esrc[e], dst = edst[e];
      float w0v = lds_w[wv][el * 128 + m];
      float w1v = lds_w[wv][el * 128 + 32 + m];
      float w2v = lds_w[wv][el * 128 + 64 + m];
      float w3v = lds_w[wv][el * 128 + 96 + m];
      float sE = xs[src * 32 + m];
      float ve0 = xv[src * 96 + m * 3 + 0];
      float ve1 = xv[src * 96 + m * 3 + 1];
      float ve2 = xv[src * 96 + m * 3 + 2];
      float y0 = Y1[e * 3 + 0], y1 = Y1[e * 3 + 1], y2 = Y1[e * 3 + 2];
      float dot = (y0 * ve0 + y1 * ve1 + y2 * ve2) * INV_SQRT3F;
      atomicAdd(&aggs[dst * 64 + m],      INV_SQRTNN * w0v * sE);
      atomicAdd(&aggs[dst * 64 + 32 + m], INV_SQRTNN * w1v * dot);
      float wsE = INV_SQRTNN * w2v * sE;
      atomicAdd(&aggv[dst * 192 + m * 3 + 0], wsE * y0);
      atomicAdd(&aggv[dst * 192 + m * 3 + 1], wsE * y1);
      atomicAdd(&aggv[dst * 192 + m * 3 + 2], wsE * y2);
      float w3s = INV_SQRTNN * w3v;
      atomicAdd(&aggv[dst * 192 + (32 + m) * 3 + 0], w3s * ve0);
      atomicAdd(&aggv[dst * 192 + (32 + m) * 3 + 1], w3s * ve1);
      atomicAdd(&aggv[dst * 192 + (32 + m) * 3 + 2], w3s * ve2);
    }
  }
}

// ---------------------------------- second FCTP on aggregates + gated update
__global__ void node_update_kernel(const float* __restrict__ aggs, const float* __restrict__ aggv,
                                   const float* __restrict__ attr,
                                   const float* __restrict__ W20, const float* __restrict__ W21,
                                   const float* __restrict__ sc_s, const float* __restrict__ sc_v,
                                   float* __restrict__ s, float* __restrict__ v,
                                   int ntiles, int N) {
  int wave = (int)(((long)blockIdx.x * blockDim.x + threadIdx.x) >> 5);
  if (wave >= ntiles) return;
  int lane = threadIdx.x & 31;
  int half = lane >> 4, lr = lane & 15;
  int node = wave * 16 + lr;
  if (node > N - 1) node = N - 1;
  float at0 = attr[node * 4 + half * 2];
  float at1 = attr[node * 4 + half * 2 + 1];

  v8f os[2] = {};
  v8f ov[3][2] = {};
  for (int k0 = 0; k0 < 256; k0 += 4) { // K = 2*MUL*NATTR = 256
    int u = k0 >> 2;
    int kr = k0 + half * 2;
    v2f b0, b1, c0, c1;
    b0.x = W20[kr * 32 + lr];        b0.y = W20[(kr + 1) * 32 + lr];
    b1.x = W20[kr * 32 + 16 + lr];   b1.y = W20[(kr + 1) * 32 + 16 + lr];
    c0.x = W21[kr * 32 + lr];        c0.y = W21[(kr + 1) * 32 + lr];
    c1.x = W21[kr * 32 + 16 + lr];   c1.y = W21[(kr + 1) * 32 + 16 + lr];
    float sv = aggs[node * 64 + u];
    v2f a; a.x = sv * at0; a.y = sv * at1;
    os[0] = wmma4(a, b0, os[0]);
    os[1] = wmma4(a, b1, os[1]);
#pragma unroll
    for (int c = 0; c < 3; c++) {
      float vv = aggv[node * 192 + u * 3 + c];
      v2f a2; a2.x = vv * at0; a2.y = vv * at1;
      ov[c][0] = wmma4(a2, c0, ov[c][0]);
      ov[c][1] = wmma4(a2, c1, ov[c][1]);
    }
  }

#pragma unroll
  for (int t = 0; t < 2; t++)
#pragma unroll
    for (int r = 0; r < 8; r++) {
      int row = wave * 16 + r + half * 8;
      if (row >= N) continue;
      int w = t * 16 + lr;
      float snew = C_S * sc_s[row * 32 + w] + C_X * (os[t][r] * INV_SQRT256);
      float sg = sigm(snew);
      s[row * 32 + w] += snew * sg; // silu
#pragma unroll
      for (int c = 0; c < 3; c++) {
        float vnew = C_S * sc_v[row * 96 + w * 3 + c] + C_X * (ov[c][t][r] * INV_SQRT256);
        v[row * 96 + w * 3 + c] += vnew * sg;
      }
    }
}

// ------------------------------------------------------- readout + pooling
__global__ void readout_kernel(const float* __restrict__ s, const float* __restrict__ attr,
                               const float* __restrict__ Wread, const int* __restrict__ batch,
                               float* __restrict__ out, int ntiles, int N, float scale) {
  int wave = (int)(((long)blockIdx.x * blockDim.x + threadIdx.x) >> 5);
  if (wave >= ntiles) return;
  int lane = threadIdx.x & 31;
  int half = lane >> 4, lr = lane & 15;
  int node = wave * 16 + lr;
  if (node > N - 1) node = N - 1;
  float at0 = attr[node * 4 + half * 2];
  float at1 = attr[node * 4 + half * 2 + 1];
  v8f acc = {};
  for (int k0 = 0; k0 < 128; k0 += 4) {
    int u = k0 >> 2;
    int kr = k0 + half * 2;
    float sv = s[node * 32 + u];
    v2f a; a.x = sv * at0; a.y = sv * at1;
    v2f b; b.x = Wread[kr * 16 + lr]; b.y = Wread[(kr + 1) * 16 + lr];
    acc = wmma4(a, b, acc);
  }
#pragma unroll
  for (int r = 0; r < 8; r++) {
    int row = wave * 16 + r + half * 8;
    if (row < N) {
      int bg = batch[row];
      atomicAdd(&out[bg * 16 + lr], acc[r] * scale);
    }
  }
}

// ------------------------------------------------------------------- launch
extern "C" void kernel_launch(void* const* d_in, const int* in_sizes, int n_in,
                              void* d_out, int out_size, void* d_ws, size_t ws_size,
                              hipStream_t stream) {
  const float* x     = (const float*)d_in[0];
  const float* attr  = (const float*)d_in[1];
  const float* evec  = (const float*)d_in[2];
  const int*   batch = (const int*)d_in[3];
  const int*   esrc  = (const int*)d_in[4];
  const int*   edst  = (const int*)d_in[5];
  const float* Wsc0  = (const float*)d_in[6];
  const float* Wsc1  = (const float*)d_in[7];
  const float* Wl10  = (const float*)d_in[8];
  const float* Wl11  = (const float*)d_in[9];
  const float* Wfc1  = (const float*)d_in[10];
  const float* Wfc2  = (const float*)d_in[11];
  const float* Wl20  = (const float*)d_in[12];
  const float* Wl21  = (const float*)d_in[13];
  const float* Wread = (const float*)d_in[14];

  int N = in_sizes[0] / 128;
  int E = in_sizes[2] / 3;
  (void)n_in; (void)ws_size;

  float* ws = (float*)d_ws;
  size_t o = 0;
  float* s    = ws + o; o += (size_t)N * 32;
  float* v    = ws + o; o += (size_t)N * 96;
  float* scs  = ws + o; o += (size_t)N * 32;
  float* scv  = ws + o; o += (size_t)N * 96;
  float* xs   = ws + o; o += (size_t)N * 32;
  float* xv   = ws + o; o += (size_t)N * 96;
  float* aggs = ws + o; o += (size_t)N * 64;
  float* aggv = ws + o; o += (size_t)N * 192; // adjacent to aggs: zeroed together
  float* Y1   = ws + o; o += (size_t)E * 3;
  float* embp = ws + o; o += (size_t)E * 12;

  int ntN = (N + 15) / 16;
  int ntE = (E + 15) / 16;

  init_sv_kernel<<<(int)(((long)N * 128 + 255) / 256), 256, 0, stream>>>(x, s, v, N);
  edge_pre_kernel<<<(E + 255) / 256, 256, 0, stream>>>(evec, Y1, embp, E);

  for (int l = 0; l < 2; l++) {
    zero_kernel<<<(int)(((long)N * 256 + 255) / 256), 256, 0, stream>>>(aggs, (long)N * 256);
    node_fctp_kernel<<<(ntN * 32 + 255) / 256, 256, 0, stream>>>(
        s, v, attr,
        Wsc0 + (size_t)l * 4096, Wsc1 + (size_t)l * 4096,
        Wl10 + (size_t)l * 4096, Wl11 + (size_t)l * 4096,
        scs, scv, xs, xv, ntN, N);
    edge_kernel<<<(ntE + EWAVES - 1) / EWAVES, EWAVES * 32, 0, stream>>>(
        embp, Y1, esrc, edst, xs, xv,
        Wfc1 + (size_t)l * 640, Wfc2 + (size_t)l * 8192,
        aggs, aggv, E, ntE);
    node_update_kernel<<<(ntN * 32 + 255) / 256, 256, 0, stream>>>(
        aggs, aggv, attr,
        Wl20 + (size_t)l * 8192, Wl21 + (size_t)l * 8192,
        scs, scv, s, v, ntN, N);
  }

  float pool_scale = INV_SQRT128 / sqrtf((float)N / 8.0f);
  zero_kernel<<<1, 256, 0, stream>>>((float*)d_out, (long)out_size);
  readout_kernel<<<(ntN * 32 + 255) / 256, 256, 0, stream>>>(
      s, attr, Wread, batch, (float*)d_out, ntN, N, pool_scale);
}